// MultiHeadAttention_90220083020022
// MI455X (gfx1250) — compile-verified
//
#include <hip/hip_runtime.h>
#include <math.h>

// MI455X / gfx1250 MHA forward, fp32 via V_WMMA_F32_16X16X4_F32,
// GEMMs staged through LDS with the Tensor Data Mover (double-buffered).
// B=2, S=2048, D=1024, H=16, dh=64.

typedef __attribute__((ext_vector_type(2))) float v2f;
typedef __attribute__((ext_vector_type(8))) float v8f;
typedef __attribute__((ext_vector_type(4))) unsigned int vu4;
typedef __attribute__((ext_vector_type(4))) int vi4;
typedef __attribute__((ext_vector_type(8))) int vi8;

static __device__ __forceinline__ v8f wmma_f32(v2f a, v2f b, v8f c) {
  // 8 args: (neg_a, A, neg_b, B, c_mod, C, reuse_a, reuse_b)
  return __builtin_amdgcn_wmma_f32_16x16x4_f32(false, a, false, b, (short)0, c,
                                               false, false);
}

// ---------------- Tensor Data Mover: 64x64 f32 tile, global -> LDS ---------
// LDS padding: 2 DWORDs after every 64 DWORDs  ->  row stride 66 floats.
#define KC 64
#define TILE_STRIDE 66
#define TILE_FLOATS (64 * TILE_STRIDE)
#define TILE_BYTES (TILE_FLOATS * 4)

static __device__ __forceinline__ void tdm_load_tile_64x64(
    unsigned lds_byte_addr, const float* gsrc, unsigned row_len_elems) {
  const unsigned long long ga = (unsigned long long)(uintptr_t)gsrc;
  // D# group 0: count=1 | lds_addr | global_addr[56:0] | type=2
  vu4 g0;
  g0[0] = 1u;
  g0[1] = lds_byte_addr;
  g0[2] = (unsigned)(ga & 0xFFFFFFFFu);
  g0[3] = (unsigned)((ga >> 32) & 0x01FFFFFFu) | (2u << 30);
  // D# group 1: data_size=2(4B), pad_enable, pad_interval=5(64dw),
  //             pad_amount=1(2dw); tensor_dim0=row_len; tensor_dim1=1<<20;
  //             tile_dim0=64, tile_dim1=64; tensor_dim0_stride=row_len.
  vi8 g1;
  g1[0] = (2 << 16) | (1 << 20) | (5 << 22) | (1 << 25);
  g1[1] = (int)((row_len_elems & 0xFFFFu) << 16);
  g1[2] = (int)((row_len_elems >> 16) & 0xFFFFu);  // tensor_dim1 low16 = 0
  g1[3] = 16 | (64 << 16);                         // tensor_dim1 hi, tile_dim0
  g1[4] = 64;                                      // tile_dim1, tile_dim2=0
  g1[5] = (int)row_len_elems;                      // dim0_stride low
  g1[6] = 0;                                       // dim0_stride hi, dim1_stride lo
  g1[7] = 0;
  vi4 z4 = {0, 0, 0, 0};
#if defined(__clang_major__) && __clang_major__ >= 23
  vi8 z8 = {0, 0, 0, 0, 0, 0, 0, 0};
  __builtin_amdgcn_tensor_load_to_lds(g0, g1, z4, z4, z8, 0);
#else
  __builtin_amdgcn_tensor_load_to_lds(g0, g1, z4, z4, 0);
#endif
}

// ---------------- GEMM: C[M,N] = A[M,K] @ Wt[N,K]^T (+ bias) ---------------
// Block = 4 waves = 64x64 C tile (each wave: 16 rows x 64 cols).
// K swept in 64-wide chunks, A/W tiles TDM-staged into LDS, double-buffered.
__global__ __launch_bounds__(128) void mha_gemm_nt_tdm(
    const float* __restrict__ A, const float* __restrict__ Wt,
    const float* __restrict__ bias, float* __restrict__ C, int M, int N,
    int K) {
  __shared__ float smem[4 * TILE_FLOATS];  // A[0],A[1],W[0],W[1]
  const int nblk = N >> 6;
  const int mb = blockIdx.x / nblk;
  const int nb = blockIdx.x % nblk;
  const int w = threadIdx.x >> 5;
  const int lane = threadIdx.x & 31;
  const int half = lane >> 4;
  const int r = lane & 15;

  const float* Abase = A + (size_t)(mb * 64) * K;
  const float* Wbase = Wt + (size_t)(nb * 64) * K;
  const unsigned smem_base = (unsigned)(uintptr_t)(void*)smem;

  if (w == 0) {  // prologue: stage chunk 0 into buffer 0
    tdm_load_tile_64x64(smem_base, Abase, (unsigned)K);
    tdm_load_tile_64x64(smem_base + 2 * TILE_BYTES, Wbase, (unsigned)K);
    __builtin_amdgcn_s_wait_tensorcnt(0);
  }
  __syncthreads();

  v8f acc[4] = {};
  for (int kc = 0; kc < K; kc += KC) {
    const int buf = (kc >> 6) & 1;
    if (w == 0 && kc + KC < K) {  // prefetch next chunk into other buffer
      const int nbuf = buf ^ 1;
      tdm_load_tile_64x64(smem_base + nbuf * TILE_BYTES, Abase + kc + KC,
                          (unsigned)K);
      tdm_load_tile_64x64(smem_base + (2 + nbuf) * TILE_BYTES, Wbase + kc + KC,
                          (unsigned)K);
    }
    const float* Ab = smem + buf * TILE_FLOATS + (16 * w + r) * TILE_STRIDE;
    const float* Wb = smem + (2 + buf) * TILE_FLOATS + r * TILE_STRIDE;
    for (int kk = 0; kk < KC; kk += 4) {
      const int ko = kk + 2 * half;
      v2f a = *(const v2f*)(Ab + ko);
      v2f b0 = *(const v2f*)(Wb + ko);
      v2f b1 = *(const v2f*)(Wb + 16 * TILE_STRIDE + ko);
      v2f b2 = *(const v2f*)(Wb + 32 * TILE_STRIDE + ko);
      v2f b3 = *(const v2f*)(Wb + 48 * TILE_STRIDE + ko);
      acc[0] = wmma_f32(a, b0, acc[0]);
      acc[1] = wmma_f32(a, b1, acc[1]);
      acc[2] = wmma_f32(a, b2, acc[2]);
      acc[3] = wmma_f32(a, b3, acc[3]);
    }
    if (w == 0) __builtin_amdgcn_s_wait_tensorcnt(0);
    __syncthreads();  // next-chunk tiles visible; this-chunk reads done
  }

  // C/D layout: VGPR i -> row (i + 8*half), col = r (within 16-wide tile t)
  for (int t = 0; t < 4; ++t) {
    const int col = nb * 64 + t * 16 + r;
    const float bv = bias ? bias[col] : 0.0f;
    for (int i = 0; i < 8; ++i) {
      const int row = mb * 64 + w * 16 + i + 8 * half;
      C[(size_t)row * N + col] = acc[t][i] + bv;
    }
  }
}

// ---------------- Flash attention, causal --------------------------------
// One wave per (b, h, 16-row q tile).  qkv: [B,S,3072], head h at cols
// [h*192, h*192+192): q=+0..63, k=+64..127, v=+128..191.  ctx: [B,S,1024].
__global__ __launch_bounds__(32) void mha_flash_attn(
    const float* __restrict__ qkv, float* __restrict__ ctx) {
  __shared__ float p_lds[256];  // 16x16 P tile: C-layout -> A-layout
  const int bid = blockIdx.x;
  const int qt = bid & 127;
  const int h = (bid >> 7) & 15;
  const int b = bid >> 11;
  const int lane = threadIdx.x & 31;
  const int half = lane >> 4;
  const int r = lane & 15;

  const float* base = qkv + (size_t)b * 2048 * 3072 + h * 192;
  const int q0 = qt * 16;

  v2f qf[16];  // Q tile as 16 A-fragments, register resident
  {
    const float* qrow = base + (size_t)(q0 + r) * 3072 + 2 * half;
    for (int c = 0; c < 16; ++c) qf[c] = *(const v2f*)(qrow + 4 * c);
  }

  v8f acc[4] = {};
  float m_run[8], l_run[8];
  for (int i = 0; i < 8; ++i) {
    m_run[i] = -1e30f;
    l_run[i] = 0.0f;
  }

  for (int k0 = 0; k0 <= q0 + 15; k0 += 16) {
    v8f s = {};
    const float* krow = base + 64 + (size_t)(k0 + r) * 3072 + 2 * half;
    for (int c = 0; c < 16; ++c) {
      v2f kf = *(const v2f*)(krow + 4 * c);
      s = wmma_f32(qf[c], kf, s);
    }
    const int kcol = k0 + r;
    float alpha[8];
    for (int i = 0; i < 8; ++i) {
      const int qabs = q0 + i + 8 * half;
      float sv = s[i] * 0.125f;  // 1/sqrt(64)
      if (kcol > qabs) sv = -INFINITY;
      float mx = sv;
      mx = fmaxf(mx, __shfl_xor(mx, 1, 32));
      mx = fmaxf(mx, __shfl_xor(mx, 2, 32));
      mx = fmaxf(mx, __shfl_xor(mx, 4, 32));
      mx = fmaxf(mx, __shfl_xor(mx, 8, 32));
      const float nm = fmaxf(m_run[i], mx);
      const float p = __expf(sv - nm);
      float rs = p;
      rs += __shfl_xor(rs, 1, 32);
      rs += __shfl_xor(rs, 2, 32);
      rs += __shfl_xor(rs, 4, 32);
      rs += __shfl_xor(rs, 8, 32);
      alpha[i] = __expf(m_run[i] - nm);
      l_run[i] = alpha[i] * l_run[i] + rs;
      m_run[i] = nm;
      p_lds[(i + 8 * half) * 16 + r] = p;
    }
    for (int t = 0; t < 4; ++t)
      for (int i = 0; i < 8; ++i) acc[t][i] = acc[t][i] * alpha[i];
    __syncthreads();  // single-wave WG: orders LDS write -> read
    for (int st = 0; st < 4; ++st) {
      const int kc = 4 * st + 2 * half;
      v2f af = *(const v2f*)(&p_lds[r * 16 + kc]);
      const float* vr = base + 128 + (size_t)(k0 + kc) * 3072 + r;
      for (int t = 0; t < 4; ++t) {
        v2f bf = {vr[t * 16], vr[t * 16 + 3072]};
        acc[t] = wmma_f32(af, bf, acc[t]);
      }
    }
    __syncthreads();
  }

  for (int t = 0; t < 4; ++t) {
    for (int i = 0; i < 8; ++i) {
      const int row = q0 + i + 8 * half;
      ctx[((size_t)b * 2048 + row) * 1024 + h * 64 + t * 16 + r] =
          acc[t][i] / l_run[i];
    }
  }
}

extern "C" void kernel_launch(void* const* d_in, const int* in_sizes, int n_in,
                              void* d_out, int out_size, void* d_ws,
                              size_t ws_size, hipStream_t stream) {
  const float* x = (const float*)d_in[0];      // [2,2048,1024]
  const float* w_qkv = (const float*)d_in[1];  // [3072,1024]
  const float* w_out = (const float*)d_in[2];  // [1024,1024]
  const float* b_out = (const float*)d_in[3];  // [1024]
  float* out = (float*)d_out;                  // [2,2048,1024]

  const int M = 2 * 2048;
  const int K = 1024;
  const int Nqkv = 3072;

  float* qkv = (float*)d_ws;            // 4096*3072 f32 = 50.3 MB
  float* ctx = qkv + (size_t)M * Nqkv;  // 4096*1024 f32 = 16.8 MB

  // QKV projection: (M/64)*(Nqkv/64) = 3072 blocks of 4 waves
  mha_gemm_nt_tdm<<<(M / 64) * (Nqkv / 64), 128, 0, stream>>>(
      x, w_qkv, nullptr, qkv, M, Nqkv, K);
  // attention: B*H*(S/16) = 4096 single-wave workgroups
  mha_flash_attn<<<2 * 16 * 128, 32, 0, stream>>>(qkv, ctx);
  // output projection + bias: (M/64)*(1024/64) = 1024 blocks
  mha_gemm_nt_tdm<<<(M / 64) * (1024 / 64), 128, 0, stream>>>(
      ctx, w_out, b_out, out, M, 1024, K);
}